// CUDARegionLoss_28905129902696
// MI455X (gfx1250) — compile-verified
//
#include <hip/hip_runtime.h>
#include <math.h>

// Problem constants (fixed by reference setup_inputs)
#define A_N   5
#define C_N   80
#define BATCH 16
#define H_N   64
#define W_N   64
#define N_GT  50
#define CH    (5 + C_N)            // 85 channels per anchor
#define HW    (H_N * W_N)          // 4096
#define CELLS (BATCH * A_N * HW)   // 327680
#define BLOCK 256
#define NBLK  (CELLS / BLOCK)      // 1280 (exact)

typedef __attribute__((ext_vector_type(2))) float v2f;
typedef __attribute__((ext_vector_type(8))) float v8f;

// Wave32 sum of one float per lane using V_WMMA_F32_16X16X4_F32.
// A = ones(16x4), B(4x16) holds {p, 0} per lane -> D[m][n] = colsum[n] for all m.
// Every B entry appears exactly once across the 64 (lane,vgpr) slots, and D row 0
// is replicated in lanes 0-15 (M=0) and 16-31 (M=8), so a full 32-lane reduce of
// D[0] yields 2 * sum(p); halve it. EXEC must be all-ones at the call site.
__device__ __forceinline__ float wave_sum32(float p) {
  v2f a; a[0] = 1.0f; a[1] = 1.0f;
  v2f b; b[0] = p;    b[1] = 0.0f;
  v8f c = {};
  v8f d = __builtin_amdgcn_wmma_f32_16x16x4_f32(false, a, false, b,
                                                (short)0, c, false, false);
  float t = d[0];
  t += __shfl_xor(t, 1, 32);
  t += __shfl_xor(t, 2, 32);
  t += __shfl_xor(t, 4, 32);
  t += __shfl_xor(t, 8, 32);
  t += __shfl_xor(t, 16, 32);
  return 0.5f * t;
}

__global__ void __launch_bounds__(BLOCK)
region_loss_main(const float* __restrict__ out,
                 const float* __restrict__ target,
                 const float* __restrict__ anchors,
                 float* __restrict__ partial) {
  // Per-batch GT boxes staged in LDS (block-uniform batch index).
  __shared__ float sx1[N_GT], sy1[N_GT], sx2[N_GT], sy2[N_GT], sarea[N_GT];
  __shared__ float scx[N_GT], scy[N_GT], sw[N_GT], sh[N_GT], scls[N_GT];
  __shared__ float swave[BLOCK / 32];

  const int tid  = threadIdx.x;
  const int cell = blockIdx.x * BLOCK + tid;
  const int b    = cell / (A_N * HW);       // uniform per block (HW % BLOCK == 0)
  const int rem  = cell % (A_N * HW);
  const int a    = rem / HW;                // uniform per block
  const int hw   = rem % HW;
  const int h    = hw / W_N;
  const int w    = hw % W_N;

  if (tid < N_GT) {
    const float* g = target + ((size_t)b * N_GT + tid) * 5;
    const float cls = g[0], cx = g[1], cy = g[2], gw = g[3], gh = g[4];
    scls[tid] = cls; scx[tid] = cx; scy[tid] = cy; sw[tid] = gw; sh[tid] = gh;
    sx1[tid] = cx - gw * 0.5f; sy1[tid] = cy - gh * 0.5f;
    sx2[tid] = cx + gw * 0.5f; sy2[tid] = cy + gh * 0.5f;
    sarea[tid] = gw * gh;
  }
  __syncthreads();

  // Channel-strided layout: ch stride = HW floats; w-contiguous per wave ->
  // each wave reads one 128B line per channel. Streaming data: non-temporal.
  const size_t base = ((size_t)(b * A_N + a) * CH) * HW + (size_t)hw;
  const float t0 = __builtin_nontemporal_load(out + base);
  const float t1 = __builtin_nontemporal_load(out + base + (size_t)HW);
  const float t2 = __builtin_nontemporal_load(out + base + (size_t)2 * HW);
  const float t3 = __builtin_nontemporal_load(out + base + (size_t)3 * HW);
  const float tc = __builtin_nontemporal_load(out + base + (size_t)4 * HW);
  __builtin_prefetch(out + base + (size_t)5 * HW, 0, 0);  // global_prefetch_b8

  const float aw = anchors[a * 2 + 0];
  const float ah = anchors[a * 2 + 1];

  // Decode predicted box (pixels).
  const float px = (1.0f / (1.0f + expf(-t0)) + (float)w) * 16.0f;
  const float py = (1.0f / (1.0f + expf(-t1)) + (float)h) * 16.0f;
  const float pw = expf(t2) * aw;
  const float ph = expf(t3) * ah;
  const float px1 = px - pw * 0.5f, py1 = py - ph * 0.5f;
  const float px2 = px + pw * 0.5f, py2 = py + ph * 0.5f;
  const float parea = pw * ph;

  // Best IoU over 50 GT boxes (strict > keeps first max, matching jnp.argmax).
  float best = -1.0f;
  int bidx = 0;
  for (int j = 0; j < N_GT; ++j) {
    const float ix1 = fmaxf(sx1[j], px1);
    const float iy1 = fmaxf(sy1[j], py1);
    const float ix2 = fminf(sx2[j], px2);
    const float iy2 = fminf(sy2[j], py2);
    const float iw = fmaxf(ix2 - ix1, 0.0f);
    const float ih = fmaxf(iy2 - iy1, 0.0f);
    const float inter = iw * ih;
    const float uni = sarea[j] + parea - inter + 1e-6f;
    const float iou = inter / uni;
    if (iou > best) { best = iou; bidx = j; }
  }

  const bool m = best > 0.6f;
  float loss;
  if (m) {
    // coord loss vs raw logits, conf loss with OBJ scale 5 -> 25*(conf-1)^2
    const float d0 = t0 - scx[bidx];
    const float d1 = t1 - scy[bidx];
    const float d2 = t2 - sw[bidx];
    const float d3 = t3 - sh[bidx];
    const float dc = tc - 1.0f;
    loss = d0 * d0 + d1 * d1 + d2 * d2 + d3 * d3 + 25.0f * dc * dc;

    // Class NLL via one-pass streaming log-sum-exp over 80 channels.
    const int tcls = (int)scls[bidx];
    float mx = -__builtin_inff();
    float s = 0.0f;
    float tl = 0.0f;
    const float* cp = out + base + (size_t)5 * HW;
    for (int cch = 0; cch < C_N; ++cch) {
      const float x = __builtin_nontemporal_load(cp);
      cp += HW;
      if (cch == tcls) tl = x;
      if (x > mx) { s = s * expf(mx - x) + 1.0f; mx = x; }
      else        { s += expf(x - mx); }
    }
    loss += (mx + logf(s)) - tl;
  } else {
    loss = tc * tc;  // NOOBJ: (conf*1 - 0)^2
  }

  // Block reduction: WMMA wave reduce (EXEC all-ones here), then LDS across waves.
  const float wsum = wave_sum32(loss);
  const int lane = tid & 31, wid = tid >> 5;
  if (lane == 0) swave[wid] = wsum;
  __syncthreads();
  if (tid == 0) {
    float t = 0.0f;
#pragma unroll
    for (int i = 0; i < BLOCK / 32; ++i) t += swave[i];
    partial[blockIdx.x] = t;
  }
}

__global__ void __launch_bounds__(BLOCK)
final_reduce(const float* __restrict__ partial, float* __restrict__ outv) {
  __shared__ float swave[BLOCK / 32];
  float p = 0.0f;
  for (int i = threadIdx.x; i < NBLK; i += BLOCK) p += partial[i];
  const float wsum = wave_sum32(p);  // all 8 waves fully active
  const int lane = threadIdx.x & 31, wid = threadIdx.x >> 5;
  if (lane == 0) swave[wid] = wsum;
  __syncthreads();
  if (threadIdx.x == 0) {
    float t = 0.0f;
#pragma unroll
    for (int i = 0; i < BLOCK / 32; ++i) t += swave[i];
    outv[0] = t;
  }
}

extern "C" void kernel_launch(void* const* d_in, const int* in_sizes, int n_in,
                              void* d_out, int out_size, void* d_ws, size_t ws_size,
                              hipStream_t stream) {
  (void)in_sizes; (void)n_in; (void)out_size; (void)ws_size;
  const float* output  = (const float*)d_in[0];
  const float* target  = (const float*)d_in[1];
  const float* anchors = (const float*)d_in[2];
  float* partial = (float*)d_ws;  // NBLK floats = 5 KB scratch

  region_loss_main<<<NBLK, BLOCK, 0, stream>>>(output, target, anchors, partial);
  final_reduce<<<1, BLOCK, 0, stream>>>(partial, (float*)d_out);
}